// LorentzInvariantPositionalEncoding_30932354465873
// MI455X (gfx1250) — compile-verified
//
#include <hip/hip_runtime.h>

// ---------------------------------------------------------------------------
// LorentzInvariantPositionalEncoding for MI455X (gfx1250, wave32)
//
// Outputs (concatenated in d_out):
//   [0 .. B*N*N)        out      = x + pe[:N,:]            (B,N,N) f32
//   [B*N*N .. 2*B*N*N)  lorentz  = pairwise Minkowski dist (B,N,N) f32
//
// lorentz via WMMA:  arg_ij = n_i + n_j - 2*M_ij,
//   M = (eta*X) X^T  computed as V_WMMA_F32_16X16X4_F32 (K = 4 coords).
// Memory-bound problem (~100 MB traffic -> ~4.3 us floor at 23.3 TB/s);
// WMMA removes the pairwise dot products from the VALU path entirely.
// ---------------------------------------------------------------------------

typedef float v2f __attribute__((ext_vector_type(2)));
typedef float v8f __attribute__((ext_vector_type(8)));

#define BATCH 32
#define NDIM  512
#define PE_MASK 0x3FFFF            // 512*512 - 1 (power of two)

// ---------------------------------------------------------------------------
// Kernel 1: out[b,i,j] = x[b,i,j] + pe[i,j]   (float4 vectorized, bandwidth)
// total elements = 32*512*512 = 8388608 ; /4 = 2097152 threads ; 8192 blocks
// ---------------------------------------------------------------------------
__global__ __launch_bounds__(256) void add_pe_kernel(const float* __restrict__ x,
                                                     const float* __restrict__ pe,
                                                     float* __restrict__ out) {
    const long idx = (long)blockIdx.x * blockDim.x + threadIdx.x;
    const long e4  = idx * 4;                       // flat element index
    const float4 xv = *(const float4*)(x + e4);
    const float4 pv = *(const float4*)(pe + (e4 & PE_MASK));
    float4 o;
    o.x = xv.x + pv.x;
    o.y = xv.y + pv.y;
    o.z = xv.z + pv.z;
    o.w = xv.w + pv.w;
    *(float4*)(out + e4) = o;
}

// ---------------------------------------------------------------------------
// Kernel 2: pairwise Lorentz distances, one 16x16 tile per wave32.
//
// A (16x4 f32) VGPR layout: lanes 0-15 hold M=0..15 / K=0,1 ; lanes 16-31
// hold M=0..15 / K=2,3  -> each lane holds 2 consecutive floats of its row.
// B (4x16) is the symmetric layout with N in place of M, so each lane again
// holds 2 consecutive floats of column-row j0+(lane&15): no transpose needed.
// D (16x16 f32, 8 VGPRs): lanes 0-15 -> M=v, N=lane ; lanes 16-31 -> M=v+8.
// ---------------------------------------------------------------------------
__global__ __launch_bounds__(256) void lorentz_wmma_kernel(const float* __restrict__ xc,
                                                           float* __restrict__ out) {
    const int lane = threadIdx.x & 31;
    const int wid  = blockIdx.x * 8 + (threadIdx.x >> 5);  // global wave id
    const int b    = wid >> 10;                            // / (32*32)
    const int it   = (wid >> 5) & 31;
    const int jt   = wid & 31;
    const int i0   = it << 4;
    const int j0   = jt << 4;
    const int l15  = lane & 15;
    const int half = lane >> 4;                            // 0: K=0,1 ; 1: K=2,3

    const float* __restrict__ xb = xc + (long)b * NDIM * 4;

    // Full coordinate rows (16B aligned loads); reused for WMMA operands + norms.
    const float4 ri = *(const float4*)(xb + (long)(i0 + l15) * 4);
    const float4 rj = *(const float4*)(xb + (long)(j0 + l15) * 4);

    // A-matrix operand: eta * x  (time component K=0 negated)
    v2f a;
    a.x = half ? ri.z : -ri.x;
    a.y = half ? ri.w :  ri.y;
    // B-matrix operand: plain x
    v2f bv;
    bv.x = half ? rj.z : rj.x;
    bv.y = half ? rj.w : rj.y;

    v8f c = {};
    // M_ij = sum_k eta_k * xc[i,k] * xc[j,k]   -> v_wmma_f32_16x16x4_f32
    v8f m = __builtin_amdgcn_wmma_f32_16x16x4_f32(
        /*neg_a=*/false, a, /*neg_b=*/false, bv,
        /*c_mod=*/(short)0, c, /*reuse_a=*/false, /*reuse_b=*/false);

    // Row norms n_r = x1^2 + x2^2 + x3^2 - x0^2
    const float n_a = ri.y * ri.y + ri.z * ri.z + ri.w * ri.w - ri.x * ri.x; // row i0+l15
    const float n_b = rj.y * rj.y + rj.z * rj.z + rj.w * rj.w - rj.x * rj.x; // col j0+l15

    float* __restrict__ ob = out + (long)b * NDIM * NDIM;
    const int col     = j0 + l15;
    const int rowbase = i0 + (half << 3);

#pragma unroll
    for (int v = 0; v < 8; ++v) {
        // broadcast n for row (rowbase+v): held by lane 8*half+v (in lanes 0-15)
        const float n_i = __shfl(n_a, (half << 3) + v, 32);
        const float arg = n_i + n_b - 2.0f * m[v];
        // raw v_sqrt_f32 (~1 ulp) instead of the refined libm sequence:
        // memory-bound kernel, no need for the Newton-correction epilogue.
        const float d   = (arg > 0.0f) ? __builtin_amdgcn_sqrtf(arg) : 0.0f;
        ob[(long)(rowbase + v) * NDIM + col] = d;
    }
}

// ---------------------------------------------------------------------------
// Launch: inputs are (x, x_coords, pe) in setup_inputs() order.
// ---------------------------------------------------------------------------
extern "C" void kernel_launch(void* const* d_in, const int* in_sizes, int n_in,
                              void* d_out, int out_size, void* d_ws, size_t ws_size,
                              hipStream_t stream) {
    (void)in_sizes; (void)n_in; (void)out_size; (void)d_ws; (void)ws_size;
    const float* x  = (const float*)d_in[0];    // (B, N, N) f32
    const float* xc = (const float*)d_in[1];    // (B, N, 4) f32
    const float* pe = (const float*)d_in[2];    // (5000, 512) f32

    float* out = (float*)d_out;                           // first tuple element
    float* lor = out + (size_t)BATCH * NDIM * NDIM;       // second tuple element

    // out = x + pe : 8388608 elems / 4 per thread / 256 per block = 8192 blocks
    add_pe_kernel<<<8192, 256, 0, stream>>>(x, pe, out);

    // 32 * 32 * 32 = 32768 tiles, 8 waves per block -> 4096 blocks
    lorentz_wmma_kernel<<<4096, 256, 0, stream>>>(xc, lor);
}